// GINConv_81544249081987
// MI455X (gfx1250) — compile-verified
//
#include <hip/hip_runtime.h>

typedef __attribute__((ext_vector_type(2))) float v2f;
typedef __attribute__((ext_vector_type(8))) float v8f;

#define DIM 128
#define LDST 132   // padded LDS row stride: bank = (4*row + col) % 64 -> conflict-free

// ---- CDNA5 async global->LDS path (ASYNCcnt), with sync fallback ----------
#if defined(__has_builtin)
#if __has_builtin(__builtin_amdgcn_global_load_async_to_lds_b128) && \
    __has_builtin(__builtin_amdgcn_s_wait_asynccnt)
#define USE_ASYNC_LDS 1
#endif
#endif
#ifndef USE_ASYNC_LDS
#define USE_ASYNC_LDS 0
#endif

// clang's builtin prototype uses a generic (gcc-style) 4 x i32 vector:
//   (addrspace(1) b128v* src, addrspace(3) b128v* dst, imm offset, imm cpol)
typedef int b128v __attribute__((vector_size(16)));

__device__ __forceinline__ void cp16_g2l(const float* g, float* l) {
#if USE_ASYNC_LDS
  __builtin_amdgcn_global_load_async_to_lds_b128(
      (__attribute__((address_space(1))) b128v*)g,
      (__attribute__((address_space(3))) b128v*)l, /*offset=*/0, /*cpol=*/0);
#else
  *reinterpret_cast<float4*>(l) = *reinterpret_cast<const float4*>(g);
#endif
}

__device__ __forceinline__ void lds_async_wait() {
#if USE_ASYNC_LDS
  __builtin_amdgcn_s_wait_asynccnt(0);
#endif
}

// ---------------------------------------------------------------------------
// Kernel 1: h0 = (1 + eps) * x
// ---------------------------------------------------------------------------
__global__ __launch_bounds__(256) void gin_combine_init(const float* __restrict__ x,
                                                        const float* __restrict__ eps,
                                                        float* __restrict__ h0,
                                                        int total4) {
  int i = blockIdx.x * 256 + threadIdx.x;
  if (i < total4) {
    float s = 1.0f + eps[0];
    float4 v = reinterpret_cast<const float4*>(x)[i];
    v.x *= s; v.y *= s; v.z *= s; v.w *= s;
    reinterpret_cast<float4*>(h0)[i] = v;
  }
}

// ---------------------------------------------------------------------------
// Kernel 2: SpMM scatter-add. One wave32 per edge (src/dst wave-uniform ->
// scalar loads); each lane moves 4 floats. x and h0 both fit in 192MB L2,
// so the f32 atomics resolve at L2 atomic units, not HBM.
// ---------------------------------------------------------------------------
__global__ __launch_bounds__(256) void gin_scatter(const float* __restrict__ x,
                                                   const int* __restrict__ src,
                                                   const int* __restrict__ dst,
                                                   float* __restrict__ h0, int E) {
  int gid = blockIdx.x * 256 + threadIdx.x;
  int e = gid >> 5;
  if (e >= E) return;
  int c = (gid & 31) << 2;
  const float4 v = *reinterpret_cast<const float4*>(x + (size_t)src[e] * DIM + c);
  float* o = h0 + (size_t)dst[e] * DIM + c;
  atomicAdd(o + 0, v.x);
  atomicAdd(o + 1, v.y);
  atomicAdd(o + 2, v.z);
  atomicAdd(o + 3, v.w);
}

// ---------------------------------------------------------------------------
// GEMM helpers: 16 rows x 128 cols per tile, 8 waves/block, each wave owns a
// 16x16 output tile; K=128 via 32x v_wmma_f32_16x16x4_f32.
// ---------------------------------------------------------------------------
__device__ __forceinline__ void load_weights_lds(float* Ws, const float* __restrict__ W,
                                                 int tid) {
#pragma unroll
  for (int j = 0; j < 16; ++j) {           // 128x128 floats = 4096 float4
    int i = tid + j * 256;
    int r = i >> 5;
    int c = (i & 31) << 2;
    cp16_g2l(W + r * DIM + c, &Ws[r * LDST + c]);
  }
}

__device__ __forceinline__ void load_atile_lds(float* As, const float* H, int row0,
                                               int N, int tid) {
#pragma unroll
  for (int j = 0; j < 2; ++j) {            // 16x128 floats = 512 float4
    int i = tid + j * 256;
    int r = i >> 5;
    int c = (i & 31) << 2;
    int row = row0 + r;
    if (row < N) {
      cp16_g2l(H + (size_t)row * DIM + c, &As[r * LDST + c]);
    } else {
      *reinterpret_cast<float4*>(&As[r * LDST + c]) = make_float4(0.f, 0.f, 0.f, 0.f);
    }
  }
}

// A-frag (16x4 f32): lanes 0-15 -> rows, VGPR0/1 = K,K+1; lanes 16-31 = K+2,K+3.
// B-frag (4x16 f32): VGPR0 = rows K / K+2; VGPR1 = rows K+1 / K+3.
__device__ __forceinline__ v8f wmma_tile_f32(const float* As, const float* Ws,
                                             int lid, int half, int N0) {
  v8f acc = {0.f, 0.f, 0.f, 0.f, 0.f, 0.f, 0.f, 0.f};
#pragma unroll
  for (int k = 0; k < DIM; k += 4) {
    const int kk = k + 2 * half;
    v2f a;
    a.x = As[lid * LDST + kk];
    a.y = As[lid * LDST + kk + 1];
    v2f b;
    b.x = Ws[kk * LDST + N0 + lid];
    b.y = Ws[(kk + 1) * LDST + N0 + lid];
    acc = __builtin_amdgcn_wmma_f32_16x16x4_f32(false, a, false, b, (short)0, acc,
                                                false, false);
  }
  return acc;
}

// ---------------------------------------------------------------------------
// Kernel 3 (persistent): T = h0 @ W1 + b1 (not stored); per-column sum/sumsq
// for BN batch stats. W1 loaded to LDS once per block; A-tiles double-buffered
// with async prefetch; stats accumulated in registers, one atomicAdd at end.
// ---------------------------------------------------------------------------
__global__ __launch_bounds__(256) void gin_gemm1_stats(const float* __restrict__ h0,
                                                       const float* __restrict__ W1,
                                                       const float* __restrict__ b1,
                                                       float* __restrict__ sums,
                                                       float* __restrict__ sumsq,
                                                       int N, int numTiles) {
  __shared__ float Ws[DIM * LDST];      // 67.6 KB
  __shared__ float As[2][16 * LDST];    // 2 x 8.4 KB
  const int tid = threadIdx.x;
  const int lane = tid & 31;
  const int lid = lane & 15;
  const int half = lane >> 4;
  const int N0 = (tid >> 5) << 4;

  load_weights_lds(Ws, W1, tid);
  const int tile0 = blockIdx.x;
  if (tile0 < numTiles) load_atile_lds(As[0], h0, tile0 << 4, N, tid);
  lds_async_wait();
  __syncthreads();

  const float bc = b1[N0 + lid];
  float s = 0.f, s2 = 0.f;
  int buf = 0;
  for (int tile = tile0; tile < numTiles; tile += gridDim.x) {
    int next = tile + (int)gridDim.x;
    if (next < numTiles) load_atile_lds(As[buf ^ 1], h0, next << 4, N, tid);

    v8f acc = wmma_tile_f32(As[buf], Ws, lid, half, N0);
    const int row0 = tile << 4;
#pragma unroll
    for (int r = 0; r < 8; ++r) {
      int row = row0 + r + (half << 3);
      float v = acc[r] + bc;
      if (row < N) { s += v; s2 += v * v; }
    }
    lds_async_wait();     // our prefetch landed in LDS
    __syncthreads();      // publish next buffer; all waves done with As[buf]
    buf ^= 1;
  }
  atomicAdd(&sums[N0 + lid], s);
  atomicAdd(&sumsq[N0 + lid], s2);
}

// ---------------------------------------------------------------------------
// Kernel 4: fold BN stats + gamma/beta into per-channel scale/shift.
// ---------------------------------------------------------------------------
__global__ void gin_bn_finalize(const float* __restrict__ sums,
                                const float* __restrict__ sumsq,
                                const float* __restrict__ gamma,
                                const float* __restrict__ beta,
                                float* __restrict__ scale,
                                float* __restrict__ shift, int N) {
  int c = threadIdx.x;               // 128 threads
  float inv_n = 1.0f / (float)N;
  float mu = sums[c] * inv_n;
  float var = sumsq[c] * inv_n - mu * mu;
  float inv = rsqrtf(var + 1e-5f);
  float sc = gamma[c] * inv;
  scale[c] = sc;
  shift[c] = beta[c] - mu * sc;
}

__global__ void gin_zero_stats(float* __restrict__ w) {
  w[threadIdx.x] = 0.f;
  w[threadIdx.x + 256] = 0.f;
}

// ---------------------------------------------------------------------------
// Kernel 5 (persistent, fused MLP): W1 AND W2 both resident in LDS (loaded
// once per block); per tile: GEMM1 -> BN+ReLU -> LDS H-tile -> GEMM2 ->
// in-place store over h0 (each tile's rows touched by exactly one block).
// A-tiles double-buffered with async prefetch. LDS ~160.5 KB -> 2 blocks/WGP.
// ---------------------------------------------------------------------------
__global__ __launch_bounds__(256) void gin_mlp_fused(const float* h0,
                                                     const float* __restrict__ W1,
                                                     const float* __restrict__ b1,
                                                     const float* __restrict__ scale,
                                                     const float* __restrict__ shift,
                                                     const float* __restrict__ W2,
                                                     const float* __restrict__ b2,
                                                     float* out, int N, int numTiles) {
  __shared__ float W1s[DIM * LDST];     // 67.6 KB
  __shared__ float W2s[DIM * LDST];     // 67.6 KB
  __shared__ float As[2][16 * LDST];    // 16.9 KB
  __shared__ float Hs[16 * LDST];       //  8.4 KB
  const int tid = threadIdx.x;
  const int lane = tid & 31;
  const int lid = lane & 15;
  const int half = lane >> 4;
  const int N0 = (tid >> 5) << 4;

  load_weights_lds(W1s, W1, tid);
  load_weights_lds(W2s, W2, tid);
  const int tile0 = blockIdx.x;
  if (tile0 < numTiles) load_atile_lds(As[0], h0, tile0 << 4, N, tid);
  lds_async_wait();
  __syncthreads();

  const float bc1 = b1[N0 + lid];
  const float sc  = scale[N0 + lid];
  const float sh  = shift[N0 + lid];
  const float bc2 = b2[N0 + lid];

  int buf = 0;
  for (int tile = tile0; tile < numTiles; tile += gridDim.x) {
    int next = tile + (int)gridDim.x;
    if (next < numTiles) load_atile_lds(As[buf ^ 1], h0, next << 4, N, tid);

    {   // GEMM1 + BatchNorm + ReLU -> Hs
      v8f acc = wmma_tile_f32(As[buf], W1s, lid, half, N0);
#pragma unroll
      for (int r = 0; r < 8; ++r) {
        float v = (acc[r] + bc1) * sc + sh;
        v = fmaxf(v, 0.f);
        Hs[(r + (half << 3)) * LDST + N0 + lid] = v;
      }
    }
    __syncthreads();      // publish Hs to all waves
    {   // GEMM2 -> global store (in-place over this tile's h0 rows)
      v8f acc = wmma_tile_f32(Hs, W2s, lid, half, N0);
      const int row0 = tile << 4;
#pragma unroll
      for (int r = 0; r < 8; ++r) {
        int row = row0 + r + (half << 3);
        if (row < N) out[(size_t)row * DIM + N0 + lid] = acc[r] + bc2;
      }
    }
    lds_async_wait();     // our A prefetch landed
    __syncthreads();      // publish next A buffer; Hs/As[buf] free for reuse
    buf ^= 1;
  }
}

// ---------------------------------------------------------------------------
extern "C" void kernel_launch(void* const* d_in, const int* in_sizes, int n_in,
                              void* d_out, int out_size, void* d_ws, size_t ws_size,
                              hipStream_t stream) {
  const float* x     = (const float*)d_in[0];
  const int*   src   = (const int*)d_in[1];
  const int*   dst   = (const int*)d_in[2];
  const float* eps   = (const float*)d_in[3];
  const float* W1    = (const float*)d_in[4];
  const float* b1    = (const float*)d_in[5];
  const float* gamma = (const float*)d_in[6];
  const float* beta  = (const float*)d_in[7];
  const float* W2    = (const float*)d_in[8];
  const float* b2    = (const float*)d_in[9];

  float* outf = (float*)d_out;   // doubles as the h0 accumulator (51 MB, fits L2)
  float* wsf  = (float*)d_ws;    // [0:128) sum, [128:256) sumsq, [256:384) scale, [384:512) shift

  const int N = in_sizes[0] / DIM;
  const int E = in_sizes[1];
  const int total4 = N * (DIM / 4);
  const int numTiles = (N + 15) / 16;
  const int blocksStats = numTiles < 1536 ? numTiles : 1536;
  const int blocksFused = numTiles < 768 ? numTiles : 768;

  gin_combine_init<<<(total4 + 255) / 256, 256, 0, stream>>>(x, eps, outf, total4);
  gin_scatter<<<(E * 32 + 255) / 256, 256, 0, stream>>>(x, src, dst, outf, E);
  gin_zero_stats<<<1, 256, 0, stream>>>(wsf);
  gin_gemm1_stats<<<blocksStats, 256, 0, stream>>>(outf, W1, b1, wsf, wsf + 128, N,
                                                   numTiles);
  gin_bn_finalize<<<1, 128, 0, stream>>>(wsf, wsf + 128, gamma, beta,
                                         wsf + 256, wsf + 384, N);
  gin_mlp_fused<<<blocksFused, 256, 0, stream>>>(outf, W1, b1, wsf + 256, wsf + 384,
                                                 W2, b2, outf, N, numTiles);
}